// RolloutBuffer_11373073400315
// MI455X (gfx1250) — compile-verified
//
#include <hip/hip_runtime.h>
#include <stdint.h>

#define N_ENVS  4096
#define T_STEPS 2048
#define TILE    256
#define NT      (T_STEPS / TILE)   // 8 tiles per env, processed high-t -> low-t
#define NWAVES  (TILE / 32)        // 8 wave32s per block

#define GAMMA_F  0.99f
#define LAMBDA_F 0.95f

#define AS3 __attribute__((address_space(3)))

// ---- CDNA5 async global->LDS load (ASYNCcnt-tracked) ------------------------
__device__ __forceinline__ void async_load_f32(const float* gp, float* lp) {
#if __has_builtin(__builtin_amdgcn_global_load_async_to_lds_b32)
  __builtin_amdgcn_global_load_async_to_lds_b32(
      (int*)(uintptr_t)gp, (int*)lp, /*imm_offset=*/0, /*cpol=*/0);
#else
  unsigned lds_addr = (unsigned)(uintptr_t)(AS3 char*)lp;
  unsigned long long gaddr = (unsigned long long)(uintptr_t)gp;
  asm volatile("global_load_async_to_lds_b32 %0, %1, off"
               :: "v"(lds_addr), "v"(gaddr) : "memory");
#endif
}

template <int N>
__device__ __forceinline__ void wait_asynccnt() {
#if __has_builtin(__builtin_amdgcn_s_wait_asynccnt)
  __builtin_amdgcn_s_wait_asynccnt(N);
#else
  if constexpr (N == 0) {
    asm volatile("s_wait_asynccnt 0" ::: "memory");
  } else {
    asm volatile("s_wait_asynccnt 4" ::: "memory");
  }
#endif
}

// ---- GAE reverse affine scan: block = one env, lane = timestep --------------
__global__ __launch_bounds__(TILE) void gae_scan_kernel(
    const float* __restrict__ rewards,   // [E][T]
    const float* __restrict__ values,    // [E][T+1]
    const float* __restrict__ first,     // [E][T+1]
    float* __restrict__ adv_out,         // [E][T]
    float* __restrict__ vt_out) {        // [E][T]
  __shared__ float Rs [2][TILE];   // r[t]
  __shared__ float Vs [2][TILE];   // V[t]
  __shared__ float V1s[2][TILE];   // V[t+1]
  __shared__ float F1s[2][TILE];   // first[t+1]
  __shared__ float wA[2][NWAVES];  // per-wave affine carry (A)
  __shared__ float wB[2][NWAVES];  // per-wave affine carry (B)

  const int e    = blockIdx.x;
  const int tid  = threadIdx.x;
  const int lane = tid & 31;
  const int wv   = tid >> 5;

  const float* Rrow = rewards + (size_t)e * T_STEPS;
  const float* Vrow = values  + (size_t)e * (T_STEPS + 1);
  const float* Frow = first   + (size_t)e * (T_STEPS + 1);
  float* Arow = adv_out + (size_t)e * T_STEPS;
  float* Wrow = vt_out  + (size_t)e * T_STEPS;

  auto issue = [&](int tile, int buf) {
    const int t = tile * TILE + tid;
    async_load_f32(Rrow + t,     &Rs [buf][tid]);
    async_load_f32(Vrow + t,     &Vs [buf][tid]);
    async_load_f32(Vrow + t + 1, &V1s[buf][tid]);
    async_load_f32(Frow + t + 1, &F1s[buf][tid]);
  };

  // Prime the double-buffered async pipeline (2 tiles in flight, 4 loads each).
  issue(NT - 1, 0);
  issue(NT - 2, 1);

  float carry = 0.0f;  // adv[t_end] of the tile to the right (adv[T] = 0)
  for (int k = 0; k < NT; ++k) {
    const int tile = NT - 1 - k;
    const int buf  = k & 1;

    // In-order async completion: <=4 outstanding means current tile landed.
    if (k < NT - 1) wait_asynccnt<4>();
    else            wait_asynccnt<0>();

    const float r  = Rs [buf][tid];
    const float v  = Vs [buf][tid];
    const float v1 = V1s[buf][tid];
    const float nl = 1.0f - F1s[buf][tid];

    // Element affine map: adv[t] = B + A * adv[t+1]
    float A = nl * (GAMMA_F * LAMBDA_F);
    float B = fmaf(nl * GAMMA_F, v1, r) - v;

    // Prefetch the tile needed two iterations from now into this buffer;
    // its latency hides behind the scan + barrier + stores of two tiles.
    if (tile >= 2) issue(tile - 2, buf);

    // Wave-level suffix scan (Hillis-Steele, gather from lane+off via bpermute).
    #pragma unroll
    for (int off = 1; off < 32; off <<= 1) {
      const int idx = (lane + off) << 2;
      const float A2 = __int_as_float(__builtin_amdgcn_ds_bpermute(idx, __float_as_int(A)));
      const float B2 = __int_as_float(__builtin_amdgcn_ds_bpermute(idx, __float_as_int(B)));
      if (lane + off < 32) { B = fmaf(A, B2, B); A *= A2; }
    }

    // lane 0 now holds the whole wave's composition.
    if (lane == 0) { wA[buf][wv] = A; wB[buf][wv] = B; }
    __syncthreads();

    float aw[NWAVES], bw[NWAVES];
    #pragma unroll
    for (int w = 0; w < NWAVES; ++w) { aw[w] = wA[buf][w]; bw[w] = wB[buf][w]; }

    // cin  = adv at this wave's right edge; cfull = adv at tile start (next carry).
    float cin = carry, cfull = carry;
    #pragma unroll
    for (int w = NWAVES - 1; w >= 0; --w) {
      if (w > wv) cin = fmaf(aw[w], cin, bw[w]);
      cfull = fmaf(aw[w], cfull, bw[w]);
    }

    const float adv = fmaf(A, cin, B);
    carry = cfull;  // uniform across all threads

    const int t = tile * TILE + tid;
    Arow[t] = adv;        // advantages
    Wrow[t] = v + adv;    // v_targ = V[t] + adv[t]
  }
}

extern "C" void kernel_launch(void* const* d_in, const int* in_sizes, int n_in,
                              void* d_out, int out_size, void* d_ws, size_t ws_size,
                              hipStream_t stream) {
  const float* rewards = (const float*)d_in[0];
  const float* values  = (const float*)d_in[1];
  const float* first   = (const float*)d_in[2];
  float* adv = (float*)d_out;
  float* vt  = adv + (size_t)N_ENVS * T_STEPS;
  gae_scan_kernel<<<dim3(N_ENVS), dim3(TILE), 0, stream>>>(rewards, values, first, adv, vt);
}